// AdvancedASI_67456756351033
// MI455X (gfx1250) — compile-verified
//
#include <hip/hip_runtime.h>
#include <hip/hip_bf16.h>
#include <math.h>

typedef __attribute__((ext_vector_type(2))) float v2f;
typedef __attribute__((ext_vector_type(8))) float v8f;

#define DD   1024
#define MM   20000
#define BQ   32          // B*Lq = 4*8
#define MC   128         // split-K chunk for attn@V
#define NCHUNK ((MM + MC - 1) / MC)

// ---------------------------------------------------------------------------
// Generic NT GEMM via fp32 WMMA:  C[i,j] = (sum_k A[i,k]*Bw[j,k] + bias[j])*scale
// blockIdx.z selects a "head": A += z*aHead, Bw += z*bHead, C += z*cHead.
// Block = 128 threads (4 waves); each wave computes a 32(i) x 64(j) tile as
// 8 accumulators of v_wmma_f32_16x16x4_f32: 2 A-fragments are reused across
// 4 B-fragments each k-step (0.75 loads per wmma).
// Inner loop is branch-free: out-of-range j-tiles clamp their B row to row 0
// (loads stay in-bounds; results discarded at the guarded store).
// Requires: ncols % 16 == 0, K % 4 == 0, 8-byte-aligned A/B bases, even lda/ldb.
// ---------------------------------------------------------------------------
__global__ void gemm_nt_wmma(const float* __restrict__ A, int lda, long long aHead, int rows,
                             const float* __restrict__ Bw, int ldb, long long bHead,
                             const float* __restrict__ bias,
                             float* __restrict__ C, int ldc, long long cHead,
                             int ncols, int K, float scale)
{
    int lane = threadIdx.x & 31;
    int wave = threadIdx.x >> 5;
    A  += (long long)blockIdx.z * aHead;
    Bw += (long long)blockIdx.z * bHead;
    C  += (long long)blockIdx.z * cHead;

    int itile = blockIdx.x * 32;
    int jbase = (blockIdx.y * 4 + wave) * 64;

    int l    = lane & 15;
    int koff = (lane >> 4) << 1;          // 0 or 2: which k-pair this half-wave holds

    int arow0 = itile + l;
    int arow1 = itile + 16 + l;
    if (arow0 >= rows) arow0 = rows - 1;  // clamp (stores are guarded)
    if (arow1 >= rows) arow1 = rows - 1;
    const float* __restrict__ ap0 = A + (size_t)arow0 * lda + koff;
    const float* __restrict__ ap1 = A + (size_t)arow1 * lda + koff;

    int jc0 = jbase, jc1 = jbase + 16, jc2 = jbase + 32, jc3 = jbase + 48;
    // clamp invalid tiles to column 0 so B loads remain in-bounds
    const float* __restrict__ bp0 = Bw + (size_t)((jc0 < ncols ? jc0 : 0) + l) * ldb + koff;
    const float* __restrict__ bp1 = Bw + (size_t)((jc1 < ncols ? jc1 : 0) + l) * ldb + koff;
    const float* __restrict__ bp2 = Bw + (size_t)((jc2 < ncols ? jc2 : 0) + l) * ldb + koff;
    const float* __restrict__ bp3 = Bw + (size_t)((jc3 < ncols ? jc3 : 0) + l) * ldb + koff;

    v8f z = {0.f,0.f,0.f,0.f,0.f,0.f,0.f,0.f};
    v8f acc00 = z, acc01 = z, acc02 = z, acc03 = z;
    v8f acc10 = z, acc11 = z, acc12 = z, acc13 = z;

    #pragma unroll 4
    for (int k = 0; k < K; k += 4) {
        v2f a0 = *(const v2f*)(ap0 + k);
        v2f a1 = *(const v2f*)(ap1 + k);
        v2f b0 = *(const v2f*)(bp0 + k);
        v2f b1 = *(const v2f*)(bp1 + k);
        v2f b2 = *(const v2f*)(bp2 + k);
        v2f b3 = *(const v2f*)(bp3 + k);
        acc00 = __builtin_amdgcn_wmma_f32_16x16x4_f32(false, a0, false, b0, (short)0, acc00, false, false);
        acc10 = __builtin_amdgcn_wmma_f32_16x16x4_f32(false, a1, false, b0, (short)0, acc10, false, false);
        acc01 = __builtin_amdgcn_wmma_f32_16x16x4_f32(false, a0, false, b1, (short)0, acc01, false, false);
        acc11 = __builtin_amdgcn_wmma_f32_16x16x4_f32(false, a1, false, b1, (short)0, acc11, false, false);
        acc02 = __builtin_amdgcn_wmma_f32_16x16x4_f32(false, a0, false, b2, (short)0, acc02, false, false);
        acc12 = __builtin_amdgcn_wmma_f32_16x16x4_f32(false, a1, false, b2, (short)0, acc12, false, false);
        acc03 = __builtin_amdgcn_wmma_f32_16x16x4_f32(false, a0, false, b3, (short)0, acc03, false, false);
        acc13 = __builtin_amdgcn_wmma_f32_16x16x4_f32(false, a1, false, b3, (short)0, acc13, false, false);
    }

    // C/D layout: VGPR r, lanes 0-15 -> M=r; lanes 16-31 -> M=r+8; N = lane&15
    int rbase0 = itile + ((lane >> 4) << 3);
    int rbase1 = rbase0 + 16;
    int cloc   = lane & 15;

    auto store_tile = [&](const v8f& acc, int jc, int rb) {
        if (jc >= ncols) return;
        int col = jc + cloc;
        float bv = bias ? bias[col] : 0.0f;
        #pragma unroll
        for (int r = 0; r < 8; ++r) {
            int row = rb + r;
            if (row < rows) C[(size_t)row * ldc + col] = (acc[r] + bv) * scale;
        }
    };
    store_tile(acc00, jc0, rbase0);
    store_tile(acc01, jc1, rbase0);
    store_tile(acc02, jc2, rbase0);
    store_tile(acc03, jc3, rbase0);
    store_tile(acc10, jc0, rbase1);
    store_tile(acc11, jc1, rbase1);
    store_tile(acc12, jc2, rbase1);
    store_tile(acc13, jc3, rbase1);
}

// ---------------------------------------------------------------------------
// In-place row softmax, one workgroup per row.
// ---------------------------------------------------------------------------
__global__ void softmax_rows(float* __restrict__ S, int ncols)
{
    __shared__ float red[256];
    int tid = threadIdx.x;
    float* R = S + (size_t)blockIdx.x * ncols;

    float mx = -3.4e38f;
    for (int i = tid; i < ncols; i += 256) mx = fmaxf(mx, R[i]);
    red[tid] = mx; __syncthreads();
    for (int s = 128; s > 0; s >>= 1) {
        if (tid < s) red[tid] = fmaxf(red[tid], red[tid + s]);
        __syncthreads();
    }
    mx = red[0]; __syncthreads();

    float sm = 0.f;
    for (int i = tid; i < ncols; i += 256) {
        float e = __expf(R[i] - mx);
        R[i] = e;
        sm += e;
    }
    red[tid] = sm; __syncthreads();
    for (int s = 128; s > 0; s >>= 1) {
        if (tid < s) red[tid] += red[tid + s];
        __syncthreads();
    }
    float inv = 1.0f / red[0];
    for (int i = tid; i < ncols; i += 256) R[i] *= inv;
}

// ---------------------------------------------------------------------------
// ctx = attn @ V, deterministic split-K: each (chunk, head) WG produces a
// BQ x 64 partial tile from LDS-staged attn/V chunks.
// attn: [16][BQ][MM], vp: [MM][DD] (head h uses cols h*64..h*64+63)
// part: [16][NCHUNK][BQ][64]
// ---------------------------------------------------------------------------
__global__ void ctx_partial(const float* __restrict__ attn, const float* __restrict__ vp,
                            float* __restrict__ part)
{
    int h  = blockIdx.y;
    int c  = blockIdx.x;
    int m0 = c * MC;
    int tid = threadIdx.x;

    __shared__ float At[BQ][MC];
    __shared__ __align__(16) float Vt[MC][64];

    for (int idx = tid; idx < BQ * MC; idx += 256) {
        int bq = idx / MC, k = idx % MC;
        int m = m0 + k;
        At[bq][k] = (m < MM) ? attn[((size_t)h * BQ + bq) * MM + m] : 0.0f;
    }
    for (int idx = tid; idx < MC * 16; idx += 256) {
        int r  = idx >> 4;
        int c4 = (idx & 15) << 2;
        int m  = m0 + r;
        if (m < MM) {
            const float* src = vp + (size_t)m * DD + h * 64 + c4;
            Vt[r][c4 + 0] = src[0]; Vt[r][c4 + 1] = src[1];
            Vt[r][c4 + 2] = src[2]; Vt[r][c4 + 3] = src[3];
        } else {
            Vt[r][c4 + 0] = 0.f; Vt[r][c4 + 1] = 0.f;
            Vt[r][c4 + 2] = 0.f; Vt[r][c4 + 3] = 0.f;
        }
    }
    __syncthreads();

    int bq = tid >> 3;
    int dg = (tid & 7) << 3;
    float acc[8] = {0.f, 0.f, 0.f, 0.f, 0.f, 0.f, 0.f, 0.f};
    for (int k = 0; k < MC; ++k) {
        float a = At[bq][k];
        #pragma unroll
        for (int j = 0; j < 8; ++j) acc[j] += a * Vt[k][dg + j];
    }
    float* dst = part + (((size_t)h * NCHUNK + c) * BQ + bq) * 64 + dg;
    #pragma unroll
    for (int j = 0; j < 8; ++j) dst[j] = acc[j];
}

__global__ void ctx_reduce(const float* __restrict__ part, float* __restrict__ ctx)
{
    int h   = blockIdx.x;
    int idx = blockIdx.y * 256 + threadIdx.x;   // 0..2047
    int bq  = idx >> 6;
    int d   = idx & 63;
    const float* p = part + ((size_t)h * NCHUNK * BQ + bq) * 64 + d;
    float s = 0.f;
    for (int c = 0; c < NCHUNK; ++c) s += p[(size_t)c * BQ * 64];
    ctx[(size_t)bq * DD + h * 64 + d] = s;
}

// ---------------------------------------------------------------------------
// Working-memory attention: 8 heads, dh=128, Lk=10. One thread per (bq, head).
// qpw already scaled by dh^-0.5.
// ---------------------------------------------------------------------------
__global__ void w_attn_kernel(const float* __restrict__ qpw, const float* __restrict__ kpw,
                              const float* __restrict__ vpw, float* __restrict__ wctx)
{
    int tid = threadIdx.x;        // 256 = 32 bq * 8 heads
    int bq  = tid >> 3;
    int h   = tid & 7;
    const float* qv = qpw + (size_t)bq * DD + h * 128;

    float sc[10];
    float mx = -3.4e38f;
    for (int k = 0; k < 10; ++k) {
        const float* kr = kpw + (size_t)k * DD + h * 128;
        float s = 0.f;
        for (int d = 0; d < 128; ++d) s += qv[d] * kr[d];
        sc[k] = s;
        mx = fmaxf(mx, s);
    }
    float sum = 0.f;
    for (int k = 0; k < 10; ++k) { sc[k] = __expf(sc[k] - mx); sum += sc[k]; }
    float inv = 1.0f / sum;

    float* outp = wctx + (size_t)bq * DD + h * 128;
    for (int d = 0; d < 128; ++d) {
        float o = 0.f;
        for (int k = 0; k < 10; ++k) o += sc[k] * vpw[(size_t)k * DD + h * 128 + d];
        outp[d] = o * inv;
    }
}

// ---------------------------------------------------------------------------
// Write path: scores2[b][m] = data[b] . ek[m]
// ---------------------------------------------------------------------------
__global__ void scores2_kernel(const float* __restrict__ data, const float* __restrict__ ek,
                               float* __restrict__ s2)
{
    __shared__ __align__(16) float dsh[4 * DD];
    int tid = threadIdx.x;
    for (int i = tid; i < 4 * DD; i += 256) dsh[i] = data[i];
    __syncthreads();

    int m = blockIdx.x * 256 + tid;
    if (m >= MM) return;

    const float4* er = (const float4*)(ek + (size_t)m * DD);
    const float4* d0 = (const float4*)(dsh);
    const float4* d1 = (const float4*)(dsh + DD);
    const float4* d2 = (const float4*)(dsh + 2 * DD);
    const float4* d3 = (const float4*)(dsh + 3 * DD);
    float a0 = 0.f, a1 = 0.f, a2 = 0.f, a3 = 0.f;
    for (int k = 0; k < DD / 4; ++k) {
        float4 e = er[k];
        float4 x;
        x = d0[k]; a0 += e.x * x.x + e.y * x.y + e.z * x.z + e.w * x.w;
        x = d1[k]; a1 += e.x * x.x + e.y * x.y + e.z * x.z + e.w * x.w;
        x = d2[k]; a2 += e.x * x.x + e.y * x.y + e.z * x.z + e.w * x.w;
        x = d3[k]; a3 += e.x * x.x + e.y * x.y + e.z * x.z + e.w * x.w;
    }
    s2[m]          = a0;
    s2[MM + m]     = a1;
    s2[2 * MM + m] = a2;
    s2[3 * MM + m] = a3;
}

__global__ void rowstat_kernel(const float* __restrict__ s2, float* __restrict__ stat)
{
    __shared__ float red[256];
    int b = blockIdx.x, tid = threadIdx.x;
    const float* S = s2 + (size_t)b * MM;

    float mx = -3.4e38f;
    for (int i = tid; i < MM; i += 256) mx = fmaxf(mx, S[i]);
    red[tid] = mx; __syncthreads();
    for (int s = 128; s > 0; s >>= 1) {
        if (tid < s) red[tid] = fmaxf(red[tid], red[tid + s]);
        __syncthreads();
    }
    mx = red[0]; __syncthreads();

    float sm = 0.f;
    for (int i = tid; i < MM; i += 256) sm += __expf(S[i] - mx);
    red[tid] = sm; __syncthreads();
    for (int s = 128; s > 0; s >>= 1) {
        if (tid < s) red[tid] += red[tid + s];
        __syncthreads();
    }
    if (tid == 0) { stat[b] = mx; stat[4 + b] = red[0]; }
}

__global__ void u_eu_kernel(const float* __restrict__ s2, const float* __restrict__ stat,
                            const float* __restrict__ eu,
                            float* __restrict__ u, float* __restrict__ neu)
{
    int m = blockIdx.x * 256 + threadIdx.x;
    if (m >= MM) return;
    float ps = 0.f;
    #pragma unroll
    for (int b = 0; b < 4; ++b)
        ps += __expf(s2[(size_t)b * MM + m] - stat[b]) / stat[4 + b];
    u[m]   = 0.0125f * ps;            // 0.05 (plast*importance) * mean over B=4
    neu[m] = eu[m] * 0.99f + ps;      // probs.sum(axis=0)
}

__global__ void dmean_kernel(const float* __restrict__ data, float* __restrict__ dmean)
{
    for (int i = threadIdx.x; i < DD; i += 256)
        dmean[i] = 0.25f * (data[i] + data[DD + i] + data[2 * DD + i] + data[3 * DD + i]);
}

__global__ void apply_kernel(const float* __restrict__ ek, const float* __restrict__ ev,
                             const float* __restrict__ u, const float* __restrict__ dmean,
                             float* __restrict__ nek, float* __restrict__ nev)
{
    size_t row = blockIdx.x;
    float um = u[row];
    float om = 1.0f - um;
    size_t off = row * DD + (size_t)threadIdx.x * 4;
    float4 e  = *(const float4*)(ek + off);
    float4 v  = *(const float4*)(ev + off);
    float4 dm = *(const float4*)(dmean + (size_t)threadIdx.x * 4);
    float4 oe, ov;
    oe.x = e.x * om + dm.x * um;  oe.y = e.y * om + dm.y * um;
    oe.z = e.z * om + dm.z * um;  oe.w = e.w * om + dm.w * um;
    ov.x = v.x * om + dm.x * um;  ov.y = v.y * om + dm.y * um;
    ov.z = v.z * om + dm.z * um;  ov.w = v.w * om + dm.w * um;
    *(float4*)(nek + off) = oe;
    *(float4*)(nev + off) = ov;
}

// ---------------------------------------------------------------------------
extern "C" void kernel_launch(void* const* d_in, const int* in_sizes, int n_in,
                              void* d_out, int out_size, void* d_ws, size_t ws_size,
                              hipStream_t stream)
{
    (void)in_sizes; (void)n_in; (void)out_size; (void)ws_size;

    const float* q    = (const float*)d_in[0];
    const float* data = (const float*)d_in[1];
    const float* ek   = (const float*)d_in[2];
    const float* ev   = (const float*)d_in[3];
    const float* sk   = (const float*)d_in[4];
    const float* sv   = (const float*)d_in[5];
    const float* wm   = (const float*)d_in[6];
    const float* Wi_e = (const float*)d_in[7];
    const float* bi_e = (const float*)d_in[8];
    const float* Wo_e = (const float*)d_in[9];
    const float* bo_e = (const float*)d_in[10];
    const float* Wi_s = (const float*)d_in[11];
    const float* bi_s = (const float*)d_in[12];
    const float* Wo_s = (const float*)d_in[13];
    const float* bo_s = (const float*)d_in[14];
    const float* Wi_w = (const float*)d_in[15];
    const float* bi_w = (const float*)d_in[16];
    const float* Wo_w = (const float*)d_in[17];
    const float* bo_w = (const float*)d_in[18];
    const float* Wrp  = (const float*)d_in[19];
    const float* brp  = (const float*)d_in[20];
    const float* eu   = (const float*)d_in[21];

    float* out      = (float*)d_out;
    float* out_read = out;                                  // [32][1024]
    float* out_ek   = out + (size_t)BQ * DD;                // [M][1024]
    float* out_ev   = out_ek + (size_t)MM * DD;
    float* out_eu   = out_ev + (size_t)MM * DD;             // [M]

    float* w = (float*)d_ws;
    size_t o = 0;
    float* kp_e  = w + o; o += (size_t)MM * DD;
    float* vp_e  = w + o; o += (size_t)MM * DD;
    float* kp_s  = w + o; o += (size_t)MM * DD;
    float* vp_s  = w + o; o += (size_t)MM * DD;
    float* sc_e  = w + o; o += (size_t)16 * BQ * MM;
    float* sc_s  = w + o; o += (size_t)16 * BQ * MM;
    float* qp_e  = w + o; o += (size_t)BQ * DD;
    float* qp_s  = w + o; o += (size_t)BQ * DD;
    float* qp_w  = w + o; o += (size_t)BQ * DD;
    float* kp_w  = w + o; o += (size_t)10 * DD;
    float* vp_w  = w + o; o += (size_t)10 * DD;
    float* e_ctx = w + o; o += (size_t)BQ * DD;
    float* s_ctx = w + o; o += (size_t)BQ * DD;
    float* w_ctx = w + o; o += (size_t)BQ * DD;
    float* cat   = w + o; o += (size_t)BQ * 3 * DD;
    float* parte = w + o; o += (size_t)16 * NCHUNK * BQ * 64;
    float* parts = w + o; o += (size_t)16 * NCHUNK * BQ * 64;
    float* s2    = w + o; o += (size_t)4 * MM;
    float* stat  = w + o; o += 8;
    float* ubuf  = w + o; o += MM;
    float* dmean = w + o; o += DD;

    auto gemm = [&](const float* A, int lda, long long aH, int rows,
                    const float* Bw, int ldb, long long bH, const float* bias,
                    float* C, int ldc, long long cH, int ncols, int K,
                    float scale, int nz) {
        dim3 g((rows + 31) / 32, (ncols + 255) / 256, nz);
        gemm_nt_wmma<<<g, dim3(128), 0, stream>>>(A, lda, aH, rows, Bw, ldb, bH,
                                                  bias, C, ldc, cH, ncols, K, scale);
    };

    const float rsq64  = 0.125f;                  // (1024/16)^-0.5
    const float rsq128 = 0.08838834764831845f;    // (1024/8)^-0.5

    // ---- q projections (scaled) ----
    gemm(q, DD, 0, BQ, Wi_e,             DD, 0, bi_e,          qp_e, DD, 0, DD, DD, rsq64, 1);
    gemm(q, DD, 0, BQ, Wi_s,             DD, 0, bi_s,          qp_s, DD, 0, DD, DD, rsq64, 1);
    gemm(q, DD, 0, BQ, Wi_w,             DD, 0, bi_w,          qp_w, DD, 0, DD, DD, rsq128, 1);

    // ---- k/v projections (the big GEMMs) ----
    gemm(ek, DD, 0, MM, Wi_e + (size_t)DD * DD,     DD, 0, bi_e + DD,     kp_e, DD, 0, DD, DD, 1.f, 1);
    gemm(ev, DD, 0, MM, Wi_e + (size_t)2 * DD * DD, DD, 0, bi_e + 2 * DD, vp_e, DD, 0, DD, DD, 1.f, 1);
    gemm(sk, DD, 0, MM, Wi_s + (size_t)DD * DD,     DD, 0, bi_s + DD,     kp_s, DD, 0, DD, DD, 1.f, 1);
    gemm(sv, DD, 0, MM, Wi_s + (size_t)2 * DD * DD, DD, 0, bi_s + 2 * DD, vp_s, DD, 0, DD, DD, 1.f, 1);
    gemm(wm, DD, 0, 10, Wi_w + (size_t)DD * DD,     DD, 0, bi_w + DD,     kp_w, DD, 0, DD, DD, 1.f, 1);
    gemm(wm, DD, 0, 10, Wi_w + (size_t)2 * DD * DD, DD, 0, bi_w + 2 * DD, vp_w, DD, 0, DD, DD, 1.f, 1);

    // ---- attention scores per head: [16] x ([32 x 64] @ [64 x 20000]) ----
    gemm(qp_e, DD, 64, BQ, kp_e, DD, 64, nullptr, sc_e, MM, (long long)BQ * MM, MM, 64, 1.f, 16);
    gemm(qp_s, DD, 64, BQ, kp_s, DD, 64, nullptr, sc_s, MM, (long long)BQ * MM, MM, 64, 1.f, 16);

    softmax_rows<<<dim3(16 * BQ), dim3(256), 0, stream>>>(sc_e, MM);
    softmax_rows<<<dim3(16 * BQ), dim3(256), 0, stream>>>(sc_s, MM);

    // ---- ctx = attn @ V (deterministic split-K) ----
    ctx_partial<<<dim3(NCHUNK, 16), dim3(256), 0, stream>>>(sc_e, vp_e, parte);
    ctx_partial<<<dim3(NCHUNK, 16), dim3(256), 0, stream>>>(sc_s, vp_s, parts);
    ctx_reduce<<<dim3(16, 8), dim3(256), 0, stream>>>(parte, e_ctx);
    ctx_reduce<<<dim3(16, 8), dim3(256), 0, stream>>>(parts, s_ctx);

    // ---- working-memory attention (tiny) ----
    w_attn_kernel<<<dim3(1), dim3(256), 0, stream>>>(qp_w, kp_w, vp_w, w_ctx);

    // ---- output projections into concat buffer, then read-projection ----
    gemm(e_ctx, DD, 0, BQ, Wo_e, DD, 0, bo_e, cat,          3 * DD, 0, DD, DD, 1.f, 1);
    gemm(s_ctx, DD, 0, BQ, Wo_s, DD, 0, bo_s, cat + DD,     3 * DD, 0, DD, DD, 1.f, 1);
    gemm(w_ctx, DD, 0, BQ, Wo_w, DD, 0, bo_w, cat + 2 * DD, 3 * DD, 0, DD, DD, 1.f, 1);
    gemm(cat, 3 * DD, 0, BQ, Wrp, 3 * DD, 0, brp, out_read, DD, 0, DD, 3 * DD, 1.f, 1);

    // ---- write path (Hebbian update) ----
    scores2_kernel<<<dim3((MM + 255) / 256), dim3(256), 0, stream>>>(data, ek, s2);
    rowstat_kernel<<<dim3(4), dim3(256), 0, stream>>>(s2, stat);
    u_eu_kernel<<<dim3((MM + 255) / 256), dim3(256), 0, stream>>>(s2, stat, eu, ubuf, out_eu);
    dmean_kernel<<<dim3(1), dim3(256), 0, stream>>>(data, dmean);
    apply_kernel<<<dim3(MM), dim3(256), 0, stream>>>(ek, ev, ubuf, dmean, out_ek, out_ev);
}